// GCN_52536039965101
// MI455X (gfx1250) — compile-verified
//
#include <hip/hip_runtime.h>

// ---------------------------------------------------------------------------
// GCN layer on MI455X (gfx1250):
//   support = x @ W                  [16384 x 64]
//   out     = relu(adj @ support + support + b)
//
// Roofline: adj (1 GiB f32) is read exactly once -> ~46us floor at 23.3 TB/s.
// f32 WMMA (V_WMMA_F32_16X16X4_F32) keeps exact f32 math at a compute time
// comparable to the memory floor, so no precision downgrade is warranted.
// adj is streamed with NT hints so it never evicts the L2-resident reused
// operands (packedB + support, 8 MiB total vs 192 MB L2).
// ---------------------------------------------------------------------------

typedef float v2f __attribute__((ext_vector_type(2)));
typedef float v8f __attribute__((ext_vector_type(8)));

#define NROW  16384
#define NFEAT 256
#define NCLS  64

// ---------------------------------------------------------------------------
// B-operand packing for V_WMMA_F32_16X16X4_F32.
// B is 4x16 (KxN). Mirroring the documented A layout, lane L holds
//   VGPR0 = B[k0][n], VGPR1 = B[k0+1][n]  with  k0 = 4*s + 2*(L/16),
//   n = ntile*16 + (L%16).
// We pre-pack so the GEMM loop does ONE coalesced b64 load per B tile:
//   packed[((s*4 + ntile)*32 + L)] = { B[k0][n], B[k0+1][n] }
// ---------------------------------------------------------------------------

// Pack W [256 x 64] -> packedW (8192 float2 = 64 steps * 4 ntiles * 32 lanes)
__global__ void gcn_packW(const float* __restrict__ W, v2f* __restrict__ pW) {
    int t = blockIdx.x * blockDim.x + threadIdx.x;   // 0..16383
    int L = t & 31;
    int n = (t >> 5) & 3;
    int s = t >> 7;                                   // K-step (K = 4*s..4*s+3)
    int k0  = 4 * s + 2 * (L >> 4);
    int col = n * 16 + (L & 15);
    v2f p;
    p.x = W[k0 * NCLS + col];
    p.y = W[(k0 + 1) * NCLS + col];
    pW[t] = p;
}

// Pack support [16384 x 64] -> packedB (524288 float2 = 4096 steps * 4 * 32)
__global__ void gcn_packSupport(const float* __restrict__ sup, v2f* __restrict__ pB) {
    int t = blockIdx.x * blockDim.x + threadIdx.x;   // 0..524287
    int L = t & 31;
    int n = (t >> 5) & 3;
    int s = t >> 7;                                   // 0..4095
    int k0  = 4 * s + 2 * (L >> 4);
    int col = n * 16 + (L & 15);
    v2f p;
    p.x = sup[k0 * NCLS + col];
    p.y = sup[(k0 + 1) * NCLS + col];
    pB[t] = p;
}

// ---------------------------------------------------------------------------
// support = x @ W.  One 16-row M-tile per block; 4 waves, wave w owns
// N-tile w (16 cols). A (x rows) is identical across the 4 waves -> L1 hit.
// ---------------------------------------------------------------------------
__global__ void gcn_support(const float* __restrict__ x,
                            const v2f*  __restrict__ pW,
                            float*      __restrict__ sup) {
    const int wave = threadIdx.x >> 5;                // N-tile 0..3
    const int L    = threadIdx.x & 31;
    const int mb   = blockIdx.x * 16;

    // A operand addressing (32-bit A 16x4): lane row = L%16, K offset 2*(L/16)
    const int arow = mb + (L & 15);
    const int koff = 2 * (L >> 4);
    const v2f* aPtr = (const v2f*)(x + (size_t)arow * NFEAT + koff);
    const v2f* bPtr = pW + wave * 32 + L;

    v8f acc = {};                                     // C starts at zero
#pragma unroll 4
    for (int s = 0; s < NFEAT / 4; ++s) {
        v2f a = aPtr[2 * s];                          // advance 4 floats / step
        v2f b = bPtr[(size_t)s * 128];                // advance 4*32 float2 / step
        acc = __builtin_amdgcn_wmma_f32_16x16x4_f32(
            /*neg_a=*/false, a, /*neg_b=*/false, b,
            /*c_mod=*/(short)0, acc, /*reuse_a=*/false, /*reuse_b=*/false);
    }

    // C/D layout: VGPR r -> row mb + r + 8*(L/16), col = wave*16 + L%16
    const int orow = mb + 8 * (L >> 4);
    const int col  = wave * 16 + (L & 15);
#pragma unroll
    for (int r = 0; r < 8; ++r)
        sup[(size_t)(orow + r) * NCLS + col] = acc[r];
}

// ---------------------------------------------------------------------------
// out = relu(adj @ support + support + b).
// Accumulator is initialized with (support + b) in C layout, so the +I term
// and bias cost nothing in the hot loop. K loop (unrolled x8): per step
// 1 NT b64 adj load (contiguous per-lane row stream) + 1 b64 packedB load
// + 1 wmma; one global_prefetch_b8 per 8 steps (each lane advances exactly
// one 128B line per 8 steps), running 2 KiB ahead of the stream.
// ---------------------------------------------------------------------------
__global__ void gcn_agg(const float* __restrict__ adj,
                        const v2f*  __restrict__ pB,
                        const float* __restrict__ sup,
                        const float* __restrict__ bias,
                        float*       __restrict__ out) {
    const int wave = threadIdx.x >> 5;                // N-tile 0..3
    const int L    = threadIdx.x & 31;
    const int mb   = blockIdx.x * 16;

    const int arow = mb + (L & 15);
    const int koff = 2 * (L >> 4);
    const v2f* aPtr = (const v2f*)(adj + (size_t)arow * NROW + koff);
    const v2f* bPtr = pB + wave * 32 + L;

    const int orow = mb + 8 * (L >> 4);
    const int col  = wave * 16 + (L & 15);
    const float bv = bias[col];

    v8f acc;
#pragma unroll
    for (int r = 0; r < 8; ++r)
        acc[r] = sup[(size_t)(orow + r) * NCLS + col] + bv;

#pragma unroll 8
    for (int s = 0; s < NROW / 4; ++s) {
        // One prefetch per cache line of the per-lane stream (16B/step x 8).
        // (s & 7) == 0 constant-folds inside the unrolled-by-8 body.
        if ((s & 7) == 0)
            __builtin_prefetch((const char*)(aPtr + 2 * (size_t)s) + 2048, 0, 0);
        // NT: adj is streamed once (1 GiB); keep it out of L2 so packedB
        // and support stay resident for all 1024 blocks.
        v2f a = __builtin_nontemporal_load(aPtr + 2 * (size_t)s);
        v2f b = bPtr[(size_t)s * 128];
        acc = __builtin_amdgcn_wmma_f32_16x16x4_f32(
            false, a, false, b, (short)0, acc, false, false);
    }

#pragma unroll
    for (int r = 0; r < 8; ++r) {
        float v = acc[r];
        float rl = v > 0.0f ? v : 0.0f;
        // Output is write-once, never re-read on device: NT store.
        __builtin_nontemporal_store(rl, &out[(size_t)(orow + r) * NCLS + col]);
    }
}

// ---------------------------------------------------------------------------
// Launch: inputs in setup_inputs() order: x, adj, W, b. Output f32 [16384,64].
// Workspace: [0,4MiB) support, [4MiB,8MiB) packedB, [8MiB,+64KiB) packedW.
// ---------------------------------------------------------------------------
extern "C" void kernel_launch(void* const* d_in, const int* in_sizes, int n_in,
                              void* d_out, int out_size, void* d_ws, size_t ws_size,
                              hipStream_t stream) {
    const float* x   = (const float*)d_in[0];
    const float* adj = (const float*)d_in[1];
    const float* W   = (const float*)d_in[2];
    const float* b   = (const float*)d_in[3];
    float* out = (float*)d_out;

    char* ws = (char*)d_ws;
    float* sup = (float*)ws;                          // 4 MiB
    v2f*   pB  = (v2f*)(ws + (4u << 20));             // 4 MiB
    v2f*   pW  = (v2f*)(ws + (8u << 20));             // 64 KiB

    gcn_packW<<<16384 / 256, 256, 0, stream>>>(W, pW);
    gcn_support<<<NROW / 16, 128, 0, stream>>>(x, pW, sup);
    gcn_packSupport<<<(NROW / 4) * 4 * 32 / 256, 256, 0, stream>>>(sup, pB);
    gcn_agg<<<NROW / 16, 128, 0, stream>>>(adj, pB, sup, b, out);
}